// MinkowskiInstanceNorm_18322330485219
// MI455X (gfx1250) — compile-verified
//
#include <hip/hip_runtime.h>

typedef float v2f __attribute__((ext_vector_type(2)));
typedef float v4f __attribute__((ext_vector_type(4)));
typedef float v8f __attribute__((ext_vector_type(8)));

#define CHANS 32
#define EPSF 1e-8f

// ---------------------------------------------------------------------------
// Workspace layout (floats):
//   ws[0            .. S*C)      : sum   -> later mean
//   ws[S*C          .. 2*S*C)    : sumsq -> later inv_std
//   ws[2*S*C        .. 2*S*C+S)  : counts
// ---------------------------------------------------------------------------

__global__ void zero_ws_kernel(float* __restrict__ ws, const int* __restrict__ nseg_p) {
    int S = *nseg_p;
    int total = 2 * S * CHANS + S;
    for (int i = threadIdx.x; i < total; i += blockDim.x) ws[i] = 0.0f;
}

// Pass 1: per-wave WMMA column reduction.  Each wave owns a contiguous row
// range; a 4-row x 32-chan chunk is reduced by 4 x V_WMMA_F32_16X16X4_F32
// (sum and sum-of-squares for each 16-channel half) with an all-ones A
// matrix, so every row of D holds the running column sums.  All control flow
// is driven by readfirstlane'd scalars so branches are SALU and EXEC stays
// all-1s around the WMMAs (ISA requirement).
__global__ __launch_bounds__(256)
void reduce_kernel(const float* __restrict__ feats,
                   const int* __restrict__ seg_ids,
                   float* __restrict__ ws,
                   const int* __restrict__ nseg_p,
                   int N, int rowsPerWave) {
    int S = __builtin_amdgcn_readfirstlane(*nseg_p);
    float* gsum = ws;
    float* gsq  = ws + (size_t)S * CHANS;
    float* gcnt = ws + (size_t)2 * S * CHANS;

    // wave id as a true SGPR -> all loop control scalar
    int waveId = __builtin_amdgcn_readfirstlane(
        (int)(blockIdx.x * (blockDim.x >> 5)) + (int)(threadIdx.x >> 5));
    int lane = threadIdx.x & 31;

    int rowBeg = waveId * rowsPerWave;
    if (rowBeg >= N) return;
    int rowEnd = rowBeg + rowsPerWave;
    if (rowEnd > N) rowEnd = N;

    int half = lane >> 4;        // which K-pair this lane carries in B
    int ch   = lane & 15;        // N index (channel within 16-chan half)
    int chan = half * 16 + ch;   // channel this lane flushes

    v2f a; a.x = 1.0f; a.y = 1.0f;          // all-ones 16x4 A matrix
    const v8f vzero = {0.f,0.f,0.f,0.f,0.f,0.f,0.f,0.f};
    v8f accS0 = vzero, accS1 = vzero;       // sums,   chans 0-15 / 16-31
    v8f accQ0 = vzero, accQ1 = vzero;       // squares
    int cnt = 0;                            // scalar rows accumulated

    int r = rowBeg;
    int cur = seg_ids[r];                   // uniform address -> s_load
    while (r < rowEnd) {
        int s0 = seg_ids[r];
        if (s0 != cur) {
            // flush accumulated stats for segment `cur` (boundaries are rare)
            float fs = half ? accS1[0] : accS0[0];
            float fq = half ? accQ1[0] : accQ0[0];
            atomicAdd(&gsum[(size_t)cur * CHANS + chan], fs);
            atomicAdd(&gsq [(size_t)cur * CHANS + chan], fq);
            if (lane == 0) atomicAdd(&gcnt[cur], (float)cnt);
            accS0 = vzero; accS1 = vzero; accQ0 = vzero; accQ1 = vzero;
            cnt = 0;
            cur = s0;
        }
        // leading run length (1..4) of rows still in segment `cur` (scalar)
        int m = 1;
        if (r + 1 < rowEnd && seg_ids[r + 1] == cur) { m = 2;
          if (r + 2 < rowEnd && seg_ids[r + 2] == cur) { m = 3;
            if (r + 3 < rowEnd && seg_ids[r + 3] == cur) m = 4; } }

        int rx = r + half * 2;           // B VGPR0 row for this lane
        int ry = rx + 1;                 // B VGPR1 row
        bool okx = (half * 2)     < m;
        bool oky = (half * 2 + 1) < m;
        size_t bx = (size_t)(rx < N ? rx : N - 1) * CHANS;
        size_t by = (size_t)(ry < N ? ry : N - 1) * CHANS;
        float x0 = __builtin_nontemporal_load(feats + bx + ch);
        float x1 = __builtin_nontemporal_load(feats + bx + ch + 16);
        float y0 = __builtin_nontemporal_load(feats + by + ch);
        float y1 = __builtin_nontemporal_load(feats + by + ch + 16);
        x0 = okx ? x0 : 0.0f;  x1 = okx ? x1 : 0.0f;
        y0 = oky ? y0 : 0.0f;  y1 = oky ? y1 : 0.0f;

        v2f b0; b0.x = x0;      b0.y = y0;       // chans 0-15
        v2f b1; b1.x = x1;      b1.y = y1;       // chans 16-31
        v2f q0; q0.x = x0 * x0; q0.y = y0 * y0;
        v2f q1; q1.x = x1 * x1; q1.y = y1 * y1;

        accS0 = __builtin_amdgcn_wmma_f32_16x16x4_f32(false, a, false, b0, (short)0, accS0, false, false);
        accS1 = __builtin_amdgcn_wmma_f32_16x16x4_f32(false, a, false, b1, (short)0, accS1, false, false);
        accQ0 = __builtin_amdgcn_wmma_f32_16x16x4_f32(false, a, false, q0, (short)0, accQ0, false, false);
        accQ1 = __builtin_amdgcn_wmma_f32_16x16x4_f32(false, a, false, q1, (short)0, accQ1, false, false);

        cnt += m;
        r   += m;
    }
    // final flush
    float fs = half ? accS1[0] : accS0[0];
    float fq = half ? accQ1[0] : accQ0[0];
    atomicAdd(&gsum[(size_t)cur * CHANS + chan], fs);
    atomicAdd(&gsq [(size_t)cur * CHANS + chan], fq);
    if (lane == 0) atomicAdd(&gcnt[cur], (float)cnt);
}

// Tiny pass: convert (sum, sumsq, count) -> (mean, inv_std) in place.
__global__ void stats_kernel(float* __restrict__ ws, const int* __restrict__ nseg_p) {
    int S = *nseg_p;
    float* gsum = ws;
    float* gsq  = ws + (size_t)S * CHANS;
    float* gcnt = ws + (size_t)2 * S * CHANS;
    for (int i = threadIdx.x; i < S * CHANS; i += blockDim.x) {
        int s = i / CHANS;
        float c = gcnt[s];
        if (c < 1.0f) c = 1.0f;
        float mean = gsum[i] / c;
        float var  = gsq[i] / c - mean * mean;
        if (var < 0.0f) var = 0.0f;
        gsum[i] = mean;
        gsq[i]  = rsqrtf(var + EPSF);
    }
}

// Pass 2: streaming normalize, 128-bit vector accesses, non-temporal on the
// 2x256MB stream so the stats/affine tables stay cached.
__global__ __launch_bounds__(256)
void norm_kernel(const float* __restrict__ feats,
                 const int* __restrict__ seg_ids,
                 const float* __restrict__ weight,
                 const float* __restrict__ bias,
                 const float* __restrict__ ws,
                 const int* __restrict__ nseg_p,
                 float* __restrict__ out, long total4) {
    int S = *nseg_p;
    const float* mean = ws;
    const float* istd = ws + (size_t)S * CHANS;
    long idx = (long)blockIdx.x * blockDim.x + threadIdx.x;
    if (idx >= total4) return;
    long row = idx >> 3;                 // CHANS/4 == 8 vec4 per row
    int  c4  = (int)(idx & 7) * 4;
    int  s   = seg_ids[row];

    v4f x  = __builtin_nontemporal_load((const v4f*)feats + idx);
    v4f mu = *(const v4f*)(mean   + (size_t)s * CHANS + c4);
    v4f iv = *(const v4f*)(istd   + (size_t)s * CHANS + c4);
    v4f w  = *(const v4f*)(weight + c4);
    v4f b  = *(const v4f*)(bias   + c4);
    v4f o  = (x - mu) * iv * w + b;
    __builtin_nontemporal_store(o, (v4f*)out + idx);
}

extern "C" void kernel_launch(void* const* d_in, const int* in_sizes, int n_in,
                              void* d_out, int out_size, void* d_ws, size_t ws_size,
                              hipStream_t stream) {
    const float* feats   = (const float*)d_in[0];
    const int*   seg_ids = (const int*)d_in[1];
    const float* weight  = (const float*)d_in[2];
    const float* bias    = (const float*)d_in[3];
    const int*   nseg_p  = (const int*)d_in[4];
    float* out = (float*)d_out;
    float* ws  = (float*)d_ws;

    int N = in_sizes[0] / CHANS;

    zero_ws_kernel<<<1, 256, 0, stream>>>(ws, nseg_p);

    const int blocks1 = 1024;                 // 8192 waves
    int wavesTotal  = blocks1 * (256 / 32);
    int rowsPerWave = (N + wavesTotal - 1) / wavesTotal;
    reduce_kernel<<<blocks1, 256, 0, stream>>>(feats, seg_ids, ws, nseg_p, N, rowsPerWave);

    stats_kernel<<<1, 256, 0, stream>>>(ws, nseg_p);

    long total4  = (long)N * (CHANS / 4);
    int  blocks2 = (int)((total4 + 255) / 256);
    norm_kernel<<<blocks2, 256, 0, stream>>>(feats, seg_ids, weight, bias, ws, nseg_p, out, total4);
}